// NetMemoryUpdater_34230889349759
// MI455X (gfx1250) — compile-verified
//
#include <hip/hip_runtime.h>
#include <math.h>

typedef float v2f __attribute__((ext_vector_type(2)));
typedef float v8f __attribute__((ext_vector_type(8)));

#define NEVENTS 100000
#define NCELLS  1024
#define EMB     128
#define NNODES  1000000

#define BPAD 132   // LDS row stride (floats) for the B tile: 132 % 64 = 4 -> conflict-free

// ---------------------------------------------------------------------------
// Squared row norms: one wave32 per row, float4 per lane, shuffle reduce.
// ---------------------------------------------------------------------------
__global__ __launch_bounds__(256) void sqnorm_kernel(const float* __restrict__ X,
                                                     float* __restrict__ out,
                                                     int nrows) {
    int wave = blockIdx.x * (blockDim.x >> 5) + (threadIdx.x >> 5);
    int lane = threadIdx.x & 31;
    if (wave >= nrows) return;
    const float4* p = (const float4*)(X + (size_t)wave * EMB);
    float4 v = p[lane];                       // 32 lanes * 4 floats = 128
    float s = v.x * v.x + v.y * v.y + v.z * v.z + v.w * v.w;
    #pragma unroll
    for (int m = 16; m >= 1; m >>= 1) s += __shfl_xor(s, m, 32);
    if (lane == 0) out[wave] = s;
}

__global__ __launch_bounds__(256) void init_keys_kernel(unsigned long long* __restrict__ keys,
                                                        int n) {
    int i = blockIdx.x * blockDim.x + threadIdx.x;
    if (i < n) keys[i] = 0xFFFFFFFFFFFFFFFFull;
}

__global__ __launch_bounds__(256) void scatter_kernel(const int* __restrict__ nodes,
                                                      const float* __restrict__ times,
                                                      float* __restrict__ ntm, int n) {
    int i = blockIdx.x * blockDim.x + threadIdx.x;
    if (i < n) ntm[nodes[i]] = times[i];
}

__global__ __launch_bounds__(256) void finalize_kernel(const unsigned long long* __restrict__ keys,
                                                       int* __restrict__ cells, int n) {
    int i = blockIdx.x * blockDim.x + threadIdx.x;
    if (i < n) cells[i] = (int)(keys[i] & 0xFFFFFFFFull);
}

// ---------------------------------------------------------------------------
// cdist GEMM with LDS-staged B:
//   Block = 8 waves, covers 256 rows x 64 cols. B tile (64 cols x 128 K,
//   32KB padded to stride 132) is loaded cooperatively into LDS once and
//   shared by all 8 waves. Each wave computes 32 rows x 64 cols using
//   V_WMMA_F32_16X16X4_F32: per K=4 step, 2 A fragments (global b64) and
//   4 B fragments (ds b64) feed 8 WMMAs -> every B fragment is used twice.
//   Epilogue: d = sqrt(max(x2 + y2 - 2*S, 0)), store, and fold argmin via
//   packed (dist_bits<<32 | col) u64 min-reduce + one atomicMin per row.
// ---------------------------------------------------------------------------
__global__ __launch_bounds__(256) void cdist_wmma_kernel(
    const float* __restrict__ A,   // embeddings [NEVENTS, EMB]
    const float* __restrict__ B,   // memory_map [NCELLS, EMB]
    const float* __restrict__ x2,  // [NEVENTS]
    const float* __restrict__ y2,  // [NCELLS]
    float* __restrict__ dists,     // [NEVENTS, NCELLS]
    unsigned long long* __restrict__ keys) // [NEVENTS]
{
    __shared__ float sB[64 * BPAD];        // 33,792 B

    const int tid  = threadIdx.x;
    const int lane = tid & 31;
    const int wave = tid >> 5;
    const int half = lane >> 4;            // 0: K pair {0,1}, 1: K pair {2,3}
    const int lmod = lane & 15;

    const int col0     = blockIdx.x * 64;
    const int blkRow0  = blockIdx.y * 256;

    // ---- Stage B tile: 64 rows x 128 floats = 2048 float4, 8 per thread ----
    #pragma unroll
    for (int i = 0; i < 8; ++i) {
        int idx = i * 256 + tid;           // [0, 2048)
        int c   = idx >> 5;                // col within tile  [0, 64)
        int kq  = idx & 31;                // float4 index in K [0, 32)
        float4 v = ((const float4*)(B + (size_t)(col0 + c) * EMB))[kq];
        *(float4*)&sB[c * BPAD + 4 * kq] = v;
    }
    __syncthreads();                       // uniform: every wave reaches this

    const int row0 = blkRow0 + wave * 32;  // this wave's 32-row slab
    if (row0 < NEVENTS) {                  // wave-uniform => EXEC all-1s for WMMA
        const float* arow0 = A + (size_t)(row0 + lmod) * EMB + 2 * half;
        const float* arow1 = arow0 + 16 * EMB;
        const float* sBf   = sB + lmod * BPAD + 2 * half;

        v8f acc[8] = {};                   // [t] rows 0-15, [t+4] rows 16-31
        #pragma unroll 2
        for (int k = 0; k < EMB; k += 4) {
            v2f a0 = *(const v2f*)(arow0 + k);
            v2f a1 = *(const v2f*)(arow1 + k);
            #pragma unroll
            for (int t = 0; t < 4; ++t) {
                v2f b = *(const v2f*)(sBf + t * 16 * BPAD + k);
                acc[t]     = __builtin_amdgcn_wmma_f32_16x16x4_f32(
                                 false, a0, false, b, (short)0, acc[t], false, false);
                acc[t + 4] = __builtin_amdgcn_wmma_f32_16x16x4_f32(
                                 false, a1, false, b, (short)0, acc[t + 4], false, false);
            }
        }

        // Epilogue. C/D layout: VGPR r, lanes 0-15 -> M=r, lanes 16-31 -> M=r+8.
        #pragma unroll
        for (int rt = 0; rt < 2; ++rt) {
            const int rbase = row0 + rt * 16;
            #pragma unroll
            for (int r = 0; r < 8; ++r) {
                const int row = rbase + r + 8 * half;
                const float xr = x2[row];
                unsigned long long best = 0xFFFFFFFFFFFFFFFFull;
                #pragma unroll
                for (int t = 0; t < 4; ++t) {
                    const int col = col0 + t * 16 + lmod;
                    float d2 = xr + y2[col] - 2.0f * acc[t + 4 * rt][r];
                    d2 = d2 > 0.0f ? d2 : 0.0f;
                    float d = sqrtf(d2);
                    dists[(size_t)row * NCELLS + col] = d;
                    unsigned long long kk =
                        ((unsigned long long)__float_as_uint(d) << 32) | (unsigned int)col;
                    best = kk < best ? kk : best;
                }
                // min-reduce across the 16 lanes sharing this row (xor 1..8 stays in half)
                #pragma unroll
                for (int m = 8; m >= 1; m >>= 1) {
                    unsigned int hi = (unsigned int)__shfl_xor((int)(best >> 32), m, 32);
                    unsigned int lo = (unsigned int)__shfl_xor((int)(best & 0xFFFFFFFFull), m, 32);
                    unsigned long long o = ((unsigned long long)hi << 32) | lo;
                    best = o < best ? o : best;
                }
                if (lmod == 0) atomicMin(keys + row, best);
            }
        }
    }
}

// ---------------------------------------------------------------------------
extern "C" void kernel_launch(void* const* d_in, const int* in_sizes, int n_in,
                              void* d_out, int out_size, void* d_ws, size_t ws_size,
                              hipStream_t stream) {
    const float* embeddings    = (const float*)d_in[0];  // [NEVENTS, EMB]
    const float* memory_map    = (const float*)d_in[1];  // [NCELLS, EMB]
    const float* times         = (const float*)d_in[2];  // [NEVENTS]
    const float* node_time_map = (const float*)d_in[3];  // [NNODES]
    const int*   nodes         = (const int*)d_in[4];    // [NEVENTS]

    float* dists = (float*)d_out;                               // [NEVENTS*NCELLS]
    int*   cells = (int*)(dists + (size_t)NEVENTS * NCELLS);    // [NEVENTS]
    float* ntm   = (float*)(cells + NEVENTS);                   // [NNODES]

    unsigned long long* keys = (unsigned long long*)d_ws;       // [NEVENTS]
    float* x2 = (float*)(keys + NEVENTS);                       // [NEVENTS]
    float* y2 = x2 + NEVENTS;                                   // [NCELLS]

    // Row/col squared norms + key init (independent, stream-ordered).
    sqnorm_kernel<<<(NEVENTS + 7) / 8, 256, 0, stream>>>(embeddings, x2, NEVENTS);
    sqnorm_kernel<<<(NCELLS + 7) / 8, 256, 0, stream>>>(memory_map, y2, NCELLS);
    init_keys_kernel<<<(NEVENTS + 255) / 256, 256, 0, stream>>>(keys, NEVENTS);

    // node_time_map: copy then indexed scatter (stream order = copy first).
    hipMemcpyAsync(ntm, node_time_map, (size_t)NNODES * sizeof(float),
                   hipMemcpyDeviceToDevice, stream);
    scatter_kernel<<<(NEVENTS + 255) / 256, 256, 0, stream>>>(nodes, times, ntm, NEVENTS);

    // Main fused cdist + argmin. Block = 8 waves; block tile 256 rows x 64 cols.
    dim3 grid(NCELLS / 64, (NEVENTS + 255) / 256);
    cdist_wmma_kernel<<<grid, 256, 0, stream>>>(embeddings, memory_map, x2, y2, dists, keys);

    finalize_kernel<<<(NEVENTS + 255) / 256, 256, 0, stream>>>(keys, cells, NEVENTS);
}